// FUSEModule_74612171866757
// MI455X (gfx1250) — compile-verified
//
#include <hip/hip_runtime.h>
#include <hip/hip_bf16.h>
#include <stdint.h>

// FUSE hydrology scan: T=8192 sequential steps over H=4096 independent channels.
// Only 128 wave32s of parallelism exist -> latency-bound, not bandwidth-bound.
// Strategy: (a) minimal per-step dependency chain (v_med3 clamps, v_log/v_exp pow,
// hand-associated FMAs so off-chain work overlaps the transcendental chain),
// (b) forcing streamed by the CDNA5 Tensor Data Mover into a 4-deep LDS ring
// (3 chunks in flight hide HBM latency behind compute), (c) non-temporal stores.

#define T_STEPS 8192
#define H_DIM   4096
#define EPS     1e-6f
#define CH      32          // timesteps per TDM chunk
#define NBUF    4           // ring depth (chunks in flight = NBUF-1)
#define LANES   32          // one wave32 per block
#define ROW_F   (LANES*3)   // 96 floats per timestep row per block
#define CHUNK_BYTES (CH*ROW_F*4)   // 12 KB per ring slot

typedef unsigned int v4u __attribute__((ext_vector_type(4)));
typedef int          v4i __attribute__((ext_vector_type(4)));
typedef int          v8i __attribute__((ext_vector_type(8)));

// Issue one TDM load: a CH x 96-float tile (row stride = H*3 floats) -> LDS.
__device__ __forceinline__ void tdm_load_chunk(const float* gsrc, unsigned lds_byte_off) {
  unsigned long long ga = (unsigned long long)(uintptr_t)gsrc;
  v4u g0;
  g0[0] = 1u;                                            // count=1 (valid user descriptor)
  g0[1] = lds_byte_off;                                  // lds_addr (bytes)
  g0[2] = (unsigned)(ga & 0xFFFFFFFFull);                // global_addr[31:0]
  g0[3] = (unsigned)((ga >> 32) & 0x01FFFFFFull)         // global_addr[56:32]
          | (2u << 30);                                  // type = 2 ("image")
  v8i g1;
  g1[0] = (int)(2u << 16);                               // data_size = 2 -> 4 bytes; wg_mask=0
  g1[1] = (int)(((unsigned)ROW_F & 0xFFFFu) << 16);      // tensor_dim0 lo16 = 96
  g1[2] = (int)(((unsigned)CH & 0xFFFFu) << 16);         // tensor_dim0 hi=0 | tensor_dim1 lo16 = CH
  g1[3] = (int)((unsigned)ROW_F << 16);                  // tensor_dim1 hi=0 | tile_dim0 = 96
  g1[4] = (int)((unsigned)CH & 0xFFFFu);                 // tile_dim1 = CH | tile_dim2 = 0
  g1[5] = (int)(H_DIM * 3);                              // tensor_dim0_stride lo32 = 12288
  g1[6] = 0;                                             // stride0 hi | stride1 lo (unused, 2D)
  g1[7] = 0;
  v4i gz = {0, 0, 0, 0};
#if defined(__clang_major__) && (__clang_major__ >= 23)
  v8i gz8 = {0, 0, 0, 0, 0, 0, 0, 0};
  __builtin_amdgcn_tensor_load_to_lds(g0, g1, gz, gz, gz8, 0);
#else
  __builtin_amdgcn_tensor_load_to_lds(g0, g1, gz, gz, 0);
#endif
}

__global__ __launch_bounds__(LANES) void fuse_scan_kernel(
    const float* __restrict__ forcing,      // (T, H, 3)
    const float* __restrict__ init_state,   // (H, 2)
    const float* __restrict__ raw_params,   // (H, 6)
    const float* __restrict__ plo,          // (6)
    const float* __restrict__ phi,          // (6)
    float* __restrict__ runoff)             // (T, H)
{
  __shared__ float smem[NBUF][CH * ROW_F];  // 48 KB ring (of 320 KB per WGP)
  const int lane = threadIdx.x;
  const int h = blockIdx.x * LANES + lane;

  // --- parameter transform (off the hot loop) ---
  float phys[6];
#pragma unroll
  for (int j = 0; j < 6; ++j) {
    float x = raw_params[h * 6 + j];
    float sig = 1.0f / (1.0f + __builtin_amdgcn_exp2f(-x * 1.4426950408889634f));
    phys[j] = plo[j] + (phi[j] - plo[j]) * sig;
  }
  const float m1 = phys[0], m2 = phys[1], percrte = phys[2];
  const float baserte = phys[3], qbp = phys[4], axv = phys[5];
  const float inv_m1 = 1.0f / m1, inv_m2 = 1.0f / m2;

  float s1 = init_state[h * 2 + 0];
  float s2 = init_state[h * 2 + 1];

  const float* gbase = forcing + (size_t)blockIdx.x * ROW_F;   // this block's column slab
  const unsigned lds_base = (unsigned)(uintptr_t)&smem[0][0];
  const size_t chunk_stride = (size_t)CH * (H_DIM * 3);        // floats between chunks

  // prime the pipeline: NBUF chunks in flight
#pragma unroll
  for (int i = 0; i < NBUF; ++i)
    tdm_load_chunk(gbase + (size_t)i * chunk_stride, lds_base + i * CHUNK_BYTES);

  const int NCHUNK = T_STEPS / CH;   // 256
  for (int c = 0; c < NCHUNK; ++c) {
    // tensor ops complete in order per wave: <=NBUF-1 outstanding => chunk c landed
    if (c + NBUF <= NCHUNK) __builtin_amdgcn_s_wait_tensorcnt(NBUF - 1);
    else                    __builtin_amdgcn_s_wait_tensorcnt(0);
    asm volatile("" ::: "memory");

    const int slot = c & (NBUF - 1);
    const float* sp = &smem[slot][lane * 3];   // stride-3 dwords: bank-conflict-free
    const size_t outb = (size_t)c * CH * H_DIM + (size_t)h;
#pragma unroll
    for (int k = 0; k < CH; ++k) {
      const float p   = sp[k * ROW_F + 0];
      const float pet = sp[k * ROW_F + 1];
      // dependency chain: mul -> v_med3 -> v_log -> mul -> v_exp -> fma -> v_med3
      const float r1  = __builtin_amdgcn_fmed3f(s1 * inv_m1, EPS, 1.0f);
      const float r2  = __builtin_amdgcn_fmed3f(s2 * inv_m2, EPS, 1.0f);
      const float pw1 = __builtin_amdgcn_exp2f(axv * __builtin_amdgcn_logf(r1));
      const float pw2 = __builtin_amdgcn_exp2f(qbp * __builtin_amdgcn_logf(r2));
      const float q12 = percrte * r1;
      const float qb  = baserte * pw2;
      // off-chain partials: ready as soon as r1/r2 are, overlap the pow chain
      const float a1  = (s1 + p) - r1 * (pet + percrte);   // s1 + p - e1 - q12
      const float a2  = s2 + q12;
      s1 = __builtin_amdgcn_fmed3f(a1 - pw1 * p, 0.0f, m1);
      s2 = __builtin_amdgcn_fmed3f(a2 - qb,      0.0f, m2);
      __builtin_nontemporal_store(pw1 * p + qb, &runoff[outb + (size_t)k * H_DIM]);
    }
    asm volatile("" ::: "memory");

    // refill the slot we just drained with chunk c+NBUF
    if (c + NBUF < NCHUNK)
      tdm_load_chunk(gbase + (size_t)(c + NBUF) * chunk_stride,
                     lds_base + slot * CHUNK_BYTES);
  }
}

extern "C" void kernel_launch(void* const* d_in, const int* in_sizes, int n_in,
                              void* d_out, int out_size, void* d_ws, size_t ws_size,
                              hipStream_t stream) {
  (void)in_sizes; (void)n_in; (void)out_size; (void)d_ws; (void)ws_size;
  const float* forcing = (const float*)d_in[0];   // (T,H,3)
  const float* init    = (const float*)d_in[1];   // (H,2)
  const float* raw     = (const float*)d_in[2];   // (H,6)
  const float* plo     = (const float*)d_in[3];   // (6)
  const float* phi     = (const float*)d_in[4];   // (6)
  float* out = (float*)d_out;                     // (T,H)

  fuse_scan_kernel<<<H_DIM / LANES, LANES, 0, stream>>>(forcing, init, raw, plo, phi, out);
}